// HierarchicalMessagePassing_79259326480428
// MI455X (gfx1250) — compile-verified
//
#include <hip/hip_runtime.h>
#include <cstddef>

// ---------------------------------------------------------------------------
// HierarchicalMessagePassing on gfx1250 (MI455X).
// Bandwidth-bound (~8 GB traffic vs ~107 GFLOP) -> keep full fp32, use
// V_WMMA_F32_16X16X4_F32. This revision reduces vmem issue pressure in the
// GEMM inner loop: W is pre-transposed into k-pair-interleaved layout so B is
// one b64 load, and each wave owns 2 row-tiles so every B load feeds 2 WMMAs.
// ---------------------------------------------------------------------------

typedef __attribute__((ext_vector_type(2))) float v2f;
typedef __attribute__((ext_vector_type(8))) float v8f;

#define UDIM 128

// One wave (32 lanes) per edge: lane i moves 4 consecutive floats.
__global__ __launch_bounds__(256) void scatter_add_kernel(
    const float* __restrict__ feat, const int* __restrict__ src,
    const int* __restrict__ dst, float* __restrict__ outfeat, int nedges) {
  const int e = blockIdx.x * 8 + (threadIdx.x >> 5);
  if (e >= nedges) return;
  const int lane = threadIdx.x & 31;
  const float4 v = *((const float4*)(feat + (size_t)src[e] * UDIM) + lane);
  float* d = outfeat + (size_t)dst[e] * UDIM + lane * 4;
  atomicAdd(d + 0, v.x);
  atomicAdd(d + 1, v.y);
  atomicAdd(d + 2, v.z);
  atomicAdd(d + 3, v.w);
}

// Wt[(k/2)*256 + col*2 + (k&1)] = W[k*128 + col]
// Makes the WMMA B-operand fetch (W[kk][col], W[kk+1][col]) one aligned b64.
__global__ __launch_bounds__(256) void transpose_w_kernel(
    const float* __restrict__ W, float* __restrict__ Wt, int total) {
  const int idx = blockIdx.x * 256 + threadIdx.x;
  if (idx >= total) return;
  const int k = idx >> 7;
  const int col = idx & 127;
  Wt[((k >> 1) << 8) + col * 2 + (k & 1)] = W[idx];
}

// out[r,:] = tanh( concat(s0[r,:], s1[r,:], (s2[r,:])) @ W + b )
// Wt is the k-pair-interleaved transpose of W ((NSEG*128) x 128 row-major).
// Each wave computes a 32-row x 128-col output stripe (2 row-tiles x 8 col
// tiles = 128 accumulator VGPRs). A wave reads only its own 32 rows of s0,
// all columns, before storing them -> in-place update (out == s0) is safe.
//
// fp32 WMMA operand layouts (CDNA5 ISA 7.12.2):
//   A (16x4):  lanes 0-15 rows M=0..15 with K=k,k+1; lanes 16-31 K=k+2,k+3
//   B (4x16):  symmetric (lane half selects K pair, column = lane&15)
//   C/D 16x16: VGPR v, lanes 0-15 -> D[v, n], lanes 16-31 -> D[v+8, n]
template <int NSEG>
__global__ __launch_bounds__(256) void gemm_tanh_kernel(
    const float* __restrict__ s0, const float* __restrict__ s1,
    const float* __restrict__ s2, const float* __restrict__ Wt,
    const float* __restrict__ bias, float* __restrict__ out, int rows) {
  const int lane = threadIdx.x & 31;
  const int wave = threadIdx.x >> 5;
  const int rowBase = (blockIdx.x * 8 + wave) * 32;
  if (rowBase >= rows) return;  // wave-uniform: EXEC stays all-ones for WMMA

  const int half = lane >> 4;  // 0: lanes 0-15, 1: lanes 16-31
  const int l15 = lane & 15;

  v8f acc0[8] = {};  // rows [rowBase,      rowBase + 16)
  v8f acc1[8] = {};  // rows [rowBase + 16, rowBase + 32)

  const float* segs[3] = {s0, s1, s2};
#pragma unroll
  for (int sgi = 0; sgi < NSEG; ++sgi) {
    const float* __restrict__ A0 = segs[sgi] + (size_t)(rowBase + l15) * UDIM;
    const float* __restrict__ A1 = A0 + 16 * UDIM;
    const float* __restrict__ Wseg = Wt + (size_t)sgi * UDIM * UDIM;
#pragma unroll 1
    for (int k = 0; k < UDIM; k += 4) {
      const int kk = k + half * 2;
      const v2f a0 = *(const v2f*)(A0 + kk);
      const v2f a1 = *(const v2f*)(A1 + kk);
      // row pair (kk, kk+1) of W, interleaved: element col at offset col*2
      const float* __restrict__ Wp = Wseg + ((size_t)(kk >> 1) << 8);
#pragma unroll
      for (int t = 0; t < 8; ++t) {
        const v2f b = *(const v2f*)(Wp + (t * 16 + l15) * 2);
        acc0[t] = __builtin_amdgcn_wmma_f32_16x16x4_f32(
            false, a0, false, b, (short)0, acc0[t], false, false);
        acc1[t] = __builtin_amdgcn_wmma_f32_16x16x4_f32(
            false, a1, false, b, (short)0, acc1[t], false, false);
      }
    }
  }

#pragma unroll
  for (int t = 0; t < 8; ++t) {
    const int col = t * 16 + l15;
    const float bv = bias[col];
    float* __restrict__ o0 = out + (size_t)(rowBase + half * 8) * UDIM + col;
    float* __restrict__ o1 = o0 + 16 * UDIM;
#pragma unroll
    for (int v = 0; v < 8; ++v) {
      o0[(size_t)v * UDIM] = tanhf(acc0[t][v] + bv);
      o1[(size_t)v * UDIM] = tanhf(acc1[t][v] + bv);
    }
  }
}

extern "C" void kernel_launch(void* const* d_in, const int* in_sizes, int n_in,
                              void* d_out, int out_size, void* d_ws,
                              size_t ws_size, hipStream_t stream) {
  (void)n_in; (void)out_size; (void)ws_size;
  const size_t N1 = 131072, N2 = 262144, N3 = 262144, N4 = 131072;

  const float* h1 = (const float*)d_in[0];
  const float* h2 = (const float*)d_in[1];
  const float* h3 = (const float*)d_in[2];
  const float* h4 = (const float*)d_in[3];
  const int* su20 = (const int*)d_in[4];  const int* du20 = (const int*)d_in[5];
  const int* su21 = (const int*)d_in[6];  const int* du21 = (const int*)d_in[7];
  const int* su30 = (const int*)d_in[8];  const int* du30 = (const int*)d_in[9];
  const int* su31 = (const int*)d_in[10]; const int* du31 = (const int*)d_in[11];
  const int* su40 = (const int*)d_in[12]; const int* du40 = (const int*)d_in[13];
  const int* su41 = (const int*)d_in[14]; const int* du41 = (const int*)d_in[15];
  const int* sd40 = (const int*)d_in[16]; const int* dd40 = (const int*)d_in[17];
  const int* sd41 = (const int*)d_in[18]; const int* dd41 = (const int*)d_in[19];
  const int* sd30 = (const int*)d_in[20]; const int* dd30 = (const int*)d_in[21];
  const int* sd31 = (const int*)d_in[22]; const int* dd31 = (const int*)d_in[23];
  const int* sd20 = (const int*)d_in[24]; const int* dd20 = (const int*)d_in[25];
  const int* sd21 = (const int*)d_in[26]; const int* dd21 = (const int*)d_in[27];
  const float* Wu2 = (const float*)d_in[28]; const float* bu2 = (const float*)d_in[29];
  const float* Wu3 = (const float*)d_in[30]; const float* bu3 = (const float*)d_in[31];
  const float* Wu4 = (const float*)d_in[32]; const float* bu4 = (const float*)d_in[33];
  const float* Wd1 = (const float*)d_in[34]; const float* bd1 = (const float*)d_in[35];
  const float* Wd2 = (const float*)d_in[36]; const float* bd2 = (const float*)d_in[37];
  const float* Wd3 = (const float*)d_in[38]; const float* bd3 = (const float*)d_in[39];

  const int Eu2 = in_sizes[4], Eu3 = in_sizes[8], Eu4 = in_sizes[12];
  const int Ed4 = in_sizes[16], Ed3 = in_sizes[20], Ed2 = in_sizes[24];

  // Output buffer doubles as working storage: [h1 | h2 | h3 | h4]
  float* out = (float*)d_out;
  float* hw1 = out;
  float* hw2 = hw1 + N1 * UDIM;
  float* hw3 = hw2 + N2 * UDIM;
  float* hw4 = hw3 + N3 * UDIM;

  // Workspace: two message buffers (2 x 128 MB) + transposed-W buffer (192 KB)
  float* buf0 = (float*)d_ws;
  float* buf1 = buf0 + N2 * UDIM;
  float* wtbuf = buf1 + N2 * UDIM;

  const size_t fb = sizeof(float) * UDIM;  // bytes per feature row
  hipMemcpyAsync(hw1, h1, N1 * fb, hipMemcpyDeviceToDevice, stream);
  hipMemcpyAsync(hw2, h2, N2 * fb, hipMemcpyDeviceToDevice, stream);
  hipMemcpyAsync(hw3, h3, N3 * fb, hipMemcpyDeviceToDevice, stream);
  hipMemcpyAsync(hw4, h4, N4 * fb, hipMemcpyDeviceToDevice, stream);

#define SCATTER(feat, s, d, buf, E) \
  scatter_add_kernel<<<((E) + 7) / 8, 256, 0, stream>>>(feat, s, d, buf, E)
#define XPOSE(W, K) \
  transpose_w_kernel<<<((K) * UDIM) / 256, 256, 0, stream>>>(W, wtbuf, (K) * UDIM)
#define GEMM3(rows, a, m0, m1, b, o) \
  gemm_tanh_kernel<3><<<(int)((rows) / 256), 256, 0, stream>>>(a, m0, m1, wtbuf, b, o, (int)(rows))
#define GEMM2(rows, a, m0, b, o) \
  gemm_tanh_kernel<2><<<(int)((rows) / 256), 256, 0, stream>>>(a, m0, nullptr, wtbuf, b, o, (int)(rows))

  // ---------------- upward pass ----------------
  hipMemsetAsync(buf0, 0, N2 * fb, stream);
  hipMemsetAsync(buf1, 0, N2 * fb, stream);
  SCATTER(hw1, su20, du20, buf0, Eu2);
  SCATTER(hw1, su21, du21, buf1, Eu2);
  XPOSE(Wu2, 3 * UDIM);
  GEMM3(N2, hw2, buf0, buf1, bu2, hw2);

  hipMemsetAsync(buf0, 0, N3 * fb, stream);
  hipMemsetAsync(buf1, 0, N3 * fb, stream);
  SCATTER(hw2, su30, du30, buf0, Eu3);
  SCATTER(hw2, su31, du31, buf1, Eu3);
  XPOSE(Wu3, 3 * UDIM);
  GEMM3(N3, hw3, buf0, buf1, bu3, hw3);

  hipMemsetAsync(buf0, 0, N4 * fb, stream);
  hipMemsetAsync(buf1, 0, N4 * fb, stream);
  SCATTER(hw3, su40, du40, buf0, Eu4);
  SCATTER(hw3, su41, du41, buf1, Eu4);
  XPOSE(Wu4, 3 * UDIM);
  GEMM3(N4, hw4, buf0, buf1, bu4, hw4);

  // ---------------- downward pass ----------------
  hipMemsetAsync(buf0, 0, N3 * fb, stream);
  SCATTER(hw4, sd40, dd40, buf0, Ed4);
  SCATTER(hw4, sd41, dd41, buf0, Ed4);
  XPOSE(Wd3, 2 * UDIM);
  GEMM2(N3, hw3, buf0, bd3, hw3);

  hipMemsetAsync(buf0, 0, N2 * fb, stream);
  SCATTER(hw3, sd30, dd30, buf0, Ed3);
  SCATTER(hw3, sd31, dd31, buf0, Ed3);
  XPOSE(Wd2, 2 * UDIM);
  GEMM2(N2, hw2, buf0, bd2, hw2);

  hipMemsetAsync(buf0, 0, N1 * fb, stream);
  SCATTER(hw2, sd20, dd20, buf0, Ed2);
  SCATTER(hw2, sd21, dd21, buf0, Ed2);
  XPOSE(Wd1, 2 * UDIM);
  GEMM2(N1, hw1, buf0, bd1, hw1);

#undef SCATTER
#undef XPOSE
#undef GEMM3
#undef GEMM2
}